// LSSFPN_81604378624479
// MI455X (gfx1250) — compile-verified
//
#include <hip/hip_runtime.h>
#include <hip/hip_bf16.h>
#include <math.h>

typedef __attribute__((ext_vector_type(2))) float v2f;
typedef __attribute__((ext_vector_type(8))) float v8f;

#define BB 2
#define NN 6
#define CC 80
#define FHh 16
#define FWw 44
#define DDd 112
#define NXx 128
#define NYy 128
#define NGRID (NYy * NXx)  // 16384
#define NPIX (FHh * FWw)   // 704
#define NCAM (BB * NN)     // 12
#define NQUAD (NPIX / 4)   // 176
#define NDG (DDd / 4)      // 28
#define CELLS_PER_BLK 64

// ---------------- small 4x4 linear algebra (device) ----------------
__device__ inline void inv4x4(const float* __restrict__ m, float* __restrict__ out) {
    float a[4][8];
    for (int i = 0; i < 4; ++i)
        for (int j = 0; j < 4; ++j) {
            a[i][j]     = m[i * 4 + j];
            a[i][j + 4] = (i == j) ? 1.0f : 0.0f;
        }
    for (int col = 0; col < 4; ++col) {
        int piv = col;
        float best = fabsf(a[col][col]);
        for (int r = col + 1; r < 4; ++r) {
            float v = fabsf(a[r][col]);
            if (v > best) { best = v; piv = r; }
        }
        if (piv != col)
            for (int j = 0; j < 8; ++j) { float t = a[col][j]; a[col][j] = a[piv][j]; a[piv][j] = t; }
        float rinv = 1.0f / a[col][col];
        for (int j = 0; j < 8; ++j) a[col][j] *= rinv;
        for (int r = 0; r < 4; ++r)
            if (r != col) {
                float f = a[r][col];
                for (int j = 0; j < 8; ++j) a[r][j] -= f * a[col][j];
            }
    }
    for (int i = 0; i < 4; ++i)
        for (int j = 0; j < 4; ++j) out[i * 4 + j] = a[i][j + 4];
}

__device__ inline void mul4x4(const float* __restrict__ a, const float* __restrict__ b,
                              float* __restrict__ c) {
    for (int i = 0; i < 4; ++i)
        for (int j = 0; j < 4; ++j) {
            float s = 0.0f;
            for (int k = 0; k < 4; ++k) s += a[i * 4 + k] * b[k * 4 + j];
            c[i * 4 + j] = s;
        }
}

// K1: per-camera combined transforms. mats[cam*32 + 0..15] = bda@s2e@inv(intrin),
//     mats[cam*32 + 16..31] = inv(ida)
__global__ void lss_prep_kernel(const float* __restrict__ s2e, const float* __restrict__ intrin,
                                const float* __restrict__ ida, const float* __restrict__ bda,
                                float* __restrict__ mats) {
    int t = threadIdx.x;
    if (t >= NCAM) return;
    int b = t / NN;
    float ii[16], tmp[16], M[16], idainv[16];
    inv4x4(intrin + t * 16, ii);
    mul4x4(s2e + t * 16, ii, tmp);
    mul4x4(bda + b * 16, tmp, M);
    inv4x4(ida + t * 16, idainv);
    for (int i = 0; i < 16; ++i) {
        mats[t * 32 + i]      = M[i];
        mats[t * 32 + 16 + i] = idainv[i];
    }
}

// K2: wave-per-pixel softmax over D=112 (wave32 butterfly reductions)
__global__ __launch_bounds__(128) void lss_softmax_kernel(const float* __restrict__ logits,
                                                          float* __restrict__ dw) {
    int wid  = (int)((blockIdx.x * blockDim.x + threadIdx.x) >> 5);
    int lane = threadIdx.x & 31;
    if (wid >= NCAM * NPIX) return;
    int cam = wid / NPIX;
    int p   = wid % NPIX;
    const float* src = logits + (size_t)cam * DDd * NPIX + p;
    float x[4];
#pragma unroll
    for (int k = 0; k < 4; ++k) {
        int d = lane + 32 * k;
        x[k] = (d < DDd) ? src[(size_t)d * NPIX] : -INFINITY;
    }
    float m = fmaxf(fmaxf(x[0], x[1]), fmaxf(x[2], x[3]));
#pragma unroll
    for (int off = 16; off >= 1; off >>= 1) m = fmaxf(m, __shfl_xor(m, off));
    float e[4];
    float s = 0.0f;
#pragma unroll
    for (int k = 0; k < 4; ++k) {
        int d = lane + 32 * k;
        e[k] = (d < DDd) ? __expf(x[k] - m) : 0.0f;
        s += e[k];
    }
#pragma unroll
    for (int off = 16; off >= 1; off >>= 1) s += __shfl_xor(s, off);
    float inv = 1.0f / s;
    float* dst = dw + (size_t)cam * DDd * NPIX + p;
#pragma unroll
    for (int k = 0; k < 4; ++k) {
        int d = lane + 32 * k;
        if (d < DDd) dst[(size_t)d * NPIX] = e[k] * inv;
    }
}

// K3: zero the channel-minor accumulator
__global__ void lss_zero_kernel(float* __restrict__ acc, int n) {
    int i = blockIdx.x * blockDim.x + threadIdx.x;
    if (i < n) acc[i] = 0.0f;
}

// K4: WMMA lift (4 pixels x 4 depth bins x 16 channels per v_wmma_f32_16x16x4_f32)
//     + coalesced atomic splat into channel-minor acc[B,NY,NX,C] (L2-resident).
//     Lanes 0-15 of each atomic hit 16 consecutive channels of one cell (64B run).
__global__ __launch_bounds__(128) void lss_splat_kernel(const float* __restrict__ ctx,
                                                        const float* __restrict__ dw,
                                                        const float* __restrict__ mats,
                                                        float* __restrict__ acc) {
    __shared__ float sm[32];
    int cam = blockIdx.y;
    int pq  = blockIdx.x;
    int b   = cam / NN;
    if (threadIdx.x < 32) sm[threadIdx.x] = mats[cam * 32 + threadIdx.x];
    __syncthreads();
    const float* M  = sm;       // bda @ s2e @ inv(intrin)
    const float* IA = sm + 16;  // inv(ida)

    int wave = threadIdx.x >> 5;
    int lane = threadIdx.x & 31;
    int mi   = lane & 15;     // this lane's A-row / output column index
    int half = lane >> 4;     // 0 or 1
    int klo  = half * 2;      // A/B K-slots owned by this lane
    int khi  = half * 2 + 1;
    int plane = mi >> 2;      // pixel-local index (0..3) for this lane's m-row
    int p0   = pq * 4;

    // B tiles: B[k,n] = ctx[pixel p0+k][ct*16+n]; 5 channel tiles held in regs
    v2f Bt[5];
    {
        int p_lo = p0 + klo;
        int p_hi = p0 + khi;
#pragma unroll
        for (int ct = 0; ct < 5; ++ct) {
            int c = ct * 16 + mi;
            Bt[ct].x = ctx[((size_t)cam * CC + c) * NPIX + p_lo];
            Bt[ct].y = ctx[((size_t)cam * CC + c) * NPIX + p_hi];
        }
    }

    int p = p0 + plane;
    int h = p / FWw, w = p % FWw;
    float u = (float)w * (703.0f / 43.0f);  // linspace(0, OGW-1, FW)
    float v = (float)h * 17.0f;             // linspace(0, OGH-1, FH)

    float* accb = acc + (size_t)b * NGRID * CC;

    for (int dg = wave; dg < NDG; dg += 4) {
        int dbin   = dg * 4 + (mi & 3);
        float dval = 2.25f + 0.5f * (float)dbin;

        // ida_inv @ (u, v, d, 1)
        float px = IA[0]  * u + IA[1]  * v + IA[2]  * dval + IA[3];
        float py = IA[4]  * u + IA[5]  * v + IA[6]  * dval + IA[7];
        float pz = IA[8]  * u + IA[9]  * v + IA[10] * dval + IA[11];
        float pw = IA[12] * u + IA[13] * v + IA[14] * dval + IA[15];
        // un-project: (x*z, y*z, z, w)
        float q0 = px * pz, q1 = py * pz, q2 = pz, q3 = pw;
        float gx = M[0] * q0 + M[1] * q1 + M[2]  * q2 + M[3]  * q3;
        float gy = M[4] * q0 + M[5] * q1 + M[6]  * q2 + M[7]  * q3;
        float gz = M[8] * q0 + M[9] * q1 + M[10] * q2 + M[11] * q3;

        // voxelize: origin - half = (-51.2, -51.2, -5.0); 1/vsize = (1.25,1.25,0.125)
        int ix = (int)floorf((gx + 51.2f) * 1.25f);
        int iy = (int)floorf((gy + 51.2f) * 1.25f);
        int iz = (int)floorf((gz + 5.0f) * 0.125f);
        bool valid = (ix >= 0) & (ix < NXx) & (iy >= 0) & (iy < NYy) & (iz == 0);

        // always-in-bounds load, then select: avoids exec-mask branch
        float wraw = dw[((size_t)cam * DDd + dbin) * NPIX + p];
        float wgt  = valid ? wraw : 0.0f;
        int cell   = valid ? (iy * NXx + ix) : -1;

        // block-diagonal A: A[m,k] = wgt(m) iff m>>2 == k
        v2f A;
        A.x = (plane == klo) ? wgt : 0.0f;
        A.y = (plane == khi) ? wgt : 0.0f;

        // cells for the 8 output rows this lane will hold (rows m = j + half*8)
        int cm[8];
#pragma unroll
        for (int j = 0; j < 8; ++j) cm[j] = __shfl(cell, j + half * 8, 32);

#pragma unroll
        for (int ct = 0; ct < 5; ++ct) {
            v8f accv = {};
            accv = __builtin_amdgcn_wmma_f32_16x16x4_f32(
                /*neg_a=*/false, A, /*neg_b=*/false, Bt[ct],
                /*c_mod=*/(short)0, accv, /*reuse_a=*/false, /*reuse_b=*/false);
            int c = ct * 16 + mi;
#pragma unroll
            for (int j = 0; j < 8; ++j) {
                if (cm[j] >= 0) {
                    // channel-minor: 16 consecutive channels per cell -> coalesced atomics
                    atomicAdd(accb + (size_t)cm[j] * CC + c, accv[j]);
                }
            }
        }
    }
}

// K5: LDS-tiled transpose acc[B,NY*NX,C] -> out[B,C,NY*NX]
__global__ __launch_bounds__(256) void lss_transpose_kernel(const float* __restrict__ acc,
                                                            float* __restrict__ out) {
    __shared__ float tile[CC][CELLS_PER_BLK + 1];  // +1 pad: avoid LDS bank conflicts
    int b     = blockIdx.y;
    int cell0 = blockIdx.x * CELLS_PER_BLK;

    const float* src = acc + ((size_t)b * NGRID + cell0) * CC;
    for (int n = threadIdx.x; n < CELLS_PER_BLK * CC; n += 256) {
        int cell = n / CC, c = n % CC;          // consecutive threads: consecutive c -> coalesced
        tile[c][cell] = src[n];
    }
    __syncthreads();

    float* dst = out + (size_t)b * CC * NGRID + cell0;
    for (int n = threadIdx.x; n < CELLS_PER_BLK * CC; n += 256) {
        int c = n / CELLS_PER_BLK, cell = n % CELLS_PER_BLK;  // consecutive cells -> coalesced
        dst[(size_t)c * NGRID + cell] = tile[c][cell];
    }
}

extern "C" void kernel_launch(void* const* d_in, const int* in_sizes, int n_in,
                              void* d_out, int out_size, void* d_ws, size_t ws_size,
                              hipStream_t stream) {
    const float* ctx    = (const float*)d_in[0];  // [B*N, C, FH, FW]
    const float* logits = (const float*)d_in[1];  // [B*N, D, FH, FW]
    const float* s2e    = (const float*)d_in[2];  // [B, N, 4, 4]
    const float* intrin = (const float*)d_in[3];  // [B, N, 4, 4]
    const float* ida    = (const float*)d_in[4];  // [B, N, 4, 4]
    const float* bda    = (const float*)d_in[5];  // [B, 4, 4]
    float* out = (float*)d_out;                   // [B, C, NY, NX]

    float* mats = (float*)d_ws;                             // 12*32 floats
    float* dw   = (float*)((char*)d_ws + 4096);             // softmaxed depth, ~3.8 MB
    float* acc  = (float*)((char*)d_ws + (4u << 20));       // channel-minor BEV, ~10.5 MB

    lss_prep_kernel<<<1, 32, 0, stream>>>(s2e, intrin, ida, bda, mats);

    int nwaves = NCAM * NPIX;  // 8448 pixels, one wave32 each
    lss_softmax_kernel<<<(nwaves * 32 + 127) / 128, 128, 0, stream>>>(logits, dw);

    int accN = BB * NGRID * CC;
    lss_zero_kernel<<<(accN + 255) / 256, 256, 0, stream>>>(acc, accN);

    dim3 grid(NQUAD, NCAM);  // 176 pixel-quads x 12 cameras
    lss_splat_kernel<<<grid, 128, 0, stream>>>(ctx, dw, mats, acc);

    dim3 tgrid(NGRID / CELLS_PER_BLK, BB);  // 256 x 2 tiles
    lss_transpose_kernel<<<tgrid, 256, 0, stream>>>(acc, out);
}